// Net_MP_11390253269724
// MI455X (gfx1250) — compile-verified
//
#include <hip/hip_runtime.h>
#include <hip/hip_bf16.h>

typedef float v2f __attribute__((ext_vector_type(2)));
typedef float v8f __attribute__((ext_vector_type(8)));

// ---------------------------------------------------------------------------
// Layer 1, node part: agg1[i][o] = x[i]@root1 + bias1   (base for edge adds)
// one lane per (node, out) pair
// ---------------------------------------------------------------------------
__global__ void __launch_bounds__(256)
node1_kernel(const float* __restrict__ x, const float* __restrict__ root1,
             const float* __restrict__ bias1, float* __restrict__ agg1, int nNodes) {
    int t = blockIdx.x * blockDim.x + threadIdx.x;
    int node = t >> 5;
    int o = t & 31;
    if (node >= nNodes) return;
    float x0 = x[node * 2 + 0];
    float x1 = x[node * 2 + 1];
    agg1[node * 32 + o] = fmaf(x0, root1[o], fmaf(x1, root1[32 + o], bias1[o]));
}

// ---------------------------------------------------------------------------
// Layer 1, edge part: msg = x[src] @ (ea0*A1 + ea1*B1 + C1); scatter-add to dst.
// 8 edges per 256-thread block, 32 lanes per edge (lane == output channel).
// ---------------------------------------------------------------------------
__global__ void __launch_bounds__(256)
edge1_kernel(const float* __restrict__ x, const int* __restrict__ ei,
             const float* __restrict__ ea, const float* __restrict__ nn1_w,
             const float* __restrict__ nn1_b, float* __restrict__ agg1, int nEdges) {
    __shared__ float sw0[32], sw1[32], sw2[32], sw3[32], sw4[32], sw5[32];
    int tid = threadIdx.x;
    if (tid < 32) {
        int o = tid;
        sw0[o] = nn1_w[o * 2 + 0];          // A1[0][o]
        sw1[o] = nn1_w[o * 2 + 1];          // B1[0][o]
        sw2[o] = nn1_b[o];                  // C1[0][o]
        sw3[o] = nn1_w[(32 + o) * 2 + 0];   // A1[1][o]
        sw4[o] = nn1_w[(32 + o) * 2 + 1];   // B1[1][o]
        sw5[o] = nn1_b[32 + o];             // C1[1][o]
    }
    __syncthreads();

    int e = blockIdx.x * 8 + (tid >> 5);
    int o = tid & 31;
    if (e >= nEdges) return;
    int src = ei[e];
    int dst = ei[nEdges + e];
    float ea0 = ea[e * 2 + 0];
    float ea1 = ea[e * 2 + 1];
    float x0 = x[src * 2 + 0];
    float x1 = x[src * 2 + 1];
    float w0 = fmaf(ea0, sw0[o], fmaf(ea1, sw1[o], sw2[o]));
    float w1 = fmaf(ea0, sw3[o], fmaf(ea1, sw4[o], sw5[o]));
    float msg = fmaf(x0, w0, x1 * w1);
    atomicAdd(&agg1[dst * 32 + o], msg);
}

// ---------------------------------------------------------------------------
// Pack Wcat[32][128] = [ A2 | B2 | C2 | root2 ] from nn2_w [1024,2], nn2_b, root2
// ---------------------------------------------------------------------------
__global__ void __launch_bounds__(256)
build_wcat_kernel(const float* __restrict__ nn2_w, const float* __restrict__ nn2_b,
                  const float* __restrict__ root2, float* __restrict__ Wcat) {
    int t = blockIdx.x * blockDim.x + threadIdx.x;
    if (t >= 1024) return;
    int i = t >> 5;
    int o = t & 31;
    int r = i * 32 + o;
    Wcat[i * 128 + 0  + o] = nn2_w[r * 2 + 0];  // A2[i][o]
    Wcat[i * 128 + 32 + o] = nn2_w[r * 2 + 1];  // B2[i][o]
    Wcat[i * 128 + 64 + o] = nn2_b[r];          // C2[i][o]
    Wcat[i * 128 + 96 + o] = root2[i * 32 + o]; // root2
}

// ---------------------------------------------------------------------------
// Dense GEMM with WMMA f32 16x16x4:
//   pqrb[N][128] = relu(agg1[N][32]) @ Wcat[32][128]   (+bias2 on cols 96..127)
// One wave per 16x16 output tile; K-loop of 8 WMMA steps.
// A 16x4 layout: lane<16 -> K={k,k+1}, lane>=16 -> K={k+2,k+3}, row = lane&15.
// B 4x16 layout: VGPR v -> row k + v + 2*(lane>=16), col = lane&15.
// D 16x16 layout: VGPR v -> row v + 8*(lane>=16), col = lane&15.
// ---------------------------------------------------------------------------
__global__ void __launch_bounds__(32)
pqrb_gemm_kernel(const float* __restrict__ agg1, const float* __restrict__ Wcat,
                 const float* __restrict__ bias2, float* __restrict__ pqrb, int nNodes) {
    const int lane = threadIdx.x;
    const int m0 = blockIdx.x * 16;
    const int n0 = blockIdx.y * 16;
    const int half = lane >> 4;           // 0 or 1
    const int l15 = lane & 15;
    int row = m0 + l15;
    if (row > nNodes - 1) row = nNodes - 1;   // clamp loads; EXEC stays all-ones
    const int col = n0 + l15;

    v8f c = {0.f, 0.f, 0.f, 0.f, 0.f, 0.f, 0.f, 0.f};
#pragma unroll
    for (int k0 = 0; k0 < 32; k0 += 4) {
        const int ka = k0 + half * 2;
        v2f a, b;
        a.x = fmaxf(agg1[row * 32 + ka + 0], 0.f);   // fused ReLU (h1)
        a.y = fmaxf(agg1[row * 32 + ka + 1], 0.f);
        b.x = Wcat[(ka + 0) * 128 + col];
        b.y = Wcat[(ka + 1) * 128 + col];
        c = __builtin_amdgcn_wmma_f32_16x16x4_f32(
                /*neg_a=*/false, a, /*neg_b=*/false, b,
                /*c_mod=*/(short)0, c, /*reuse_a=*/false, /*reuse_b=*/false);
    }

    const float badd = (col >= 96) ? bias2[col - 96] : 0.0f;
#pragma unroll
    for (int v = 0; v < 8; ++v) {
        int r = m0 + v + 8 * half;
        if (r < nNodes) pqrb[r * 128 + col] = c[v] + badd;
    }
}

// ---------------------------------------------------------------------------
// Layer 2, edge part: msg = ea0*p[src] + ea1*q[src] + r[src]; scatter-add into
// the "base" chunk (cols 96..127) of pqrb, which already holds h1@root2+bias2.
// ---------------------------------------------------------------------------
__global__ void __launch_bounds__(256)
edge2_kernel(const int* __restrict__ ei, const float* __restrict__ ea,
             float* __restrict__ pqrb, int nEdges) {
    int tid = threadIdx.x;
    int e = blockIdx.x * 8 + (tid >> 5);
    int o = tid & 31;
    if (e >= nEdges) return;
    int src = ei[e];
    int dst = ei[nEdges + e];
    float ea0 = ea[e * 2 + 0];
    float ea1 = ea[e * 2 + 1];
    const float* sp = pqrb + (size_t)src * 128;
    float msg = fmaf(ea0, sp[o], fmaf(ea1, sp[32 + o], sp[64 + o]));
    atomicAdd(&pqrb[(size_t)dst * 128 + 96 + o], msg);
}

// ---------------------------------------------------------------------------
// Head: h2 = relu(agg2); h3 = relu(h2 @ fc1_w^T + fc1_b); out = h3 @ fc2_w^T + fc2_b
// one thread per node; fc1 weights staged in LDS.
// ---------------------------------------------------------------------------
__global__ void __launch_bounds__(256)
final_kernel(const float* __restrict__ pqrb, const float* __restrict__ fc1_w,
             const float* __restrict__ fc1_b, const float* __restrict__ fc2_w,
             const float* __restrict__ fc2_b, float* __restrict__ out, int nNodes) {
    __shared__ float sW[1024];
    __shared__ float sb[32];
    __shared__ float sw2[32];
    int tid = threadIdx.x;
    for (int i = tid; i < 1024; i += 256) sW[i] = fc1_w[i];
    if (tid < 32) { sb[tid] = fc1_b[tid]; sw2[tid] = fc2_w[tid]; }
    __syncthreads();

    int node = blockIdx.x * blockDim.x + tid;
    if (node >= nNodes) return;

    float h[32];
    const float* a = pqrb + (size_t)node * 128 + 96;
#pragma unroll
    for (int o = 0; o < 32; ++o) h[o] = fmaxf(a[o], 0.0f);

    float acc = fc2_b[0];
#pragma unroll 4
    for (int p = 0; p < 32; ++p) {
        float s = sb[p];
#pragma unroll
        for (int o = 0; o < 32; ++o) s = fmaf(h[o], sW[p * 32 + o], s);
        acc = fmaf(fmaxf(s, 0.0f), sw2[p], acc);
    }
    out[node] = acc;
}

// ---------------------------------------------------------------------------
extern "C" void kernel_launch(void* const* d_in, const int* in_sizes, int n_in,
                              void* d_out, int out_size, void* d_ws, size_t ws_size,
                              hipStream_t stream) {
    const float* x      = (const float*)d_in[0];
    const int*   ei     = (const int*)  d_in[1];
    const float* ea     = (const float*)d_in[2];
    const float* nn1_w  = (const float*)d_in[3];
    const float* nn1_b  = (const float*)d_in[4];
    const float* root1  = (const float*)d_in[5];
    const float* bias1  = (const float*)d_in[6];
    const float* nn2_w  = (const float*)d_in[7];
    const float* nn2_b  = (const float*)d_in[8];
    const float* root2  = (const float*)d_in[9];
    const float* bias2  = (const float*)d_in[10];
    const float* fc1_w  = (const float*)d_in[11];
    const float* fc1_b  = (const float*)d_in[12];
    const float* fc2_w  = (const float*)d_in[13];
    const float* fc2_b  = (const float*)d_in[14];
    float* out = (float*)d_out;

    const int nNodes = in_sizes[0] / 2;   // x is [N,2]
    const int nEdges = in_sizes[1] / 2;   // edge_index is [2,E]

    // Workspace layout: agg1 [N*32] | pqrb [N*128] | Wcat [32*128]
    float* agg1 = (float*)d_ws;
    float* pqrb = agg1 + (size_t)nNodes * 32;
    float* Wcat = pqrb + (size_t)nNodes * 128;

    // Layer 1 base (x@root1+bias1) directly into the accumulator
    node1_kernel<<<(nNodes * 32 + 255) / 256, 256, 0, stream>>>(x, root1, bias1, agg1, nNodes);
    // Pack layer-2 weights [A2|B2|C2|root2]
    build_wcat_kernel<<<4, 256, 0, stream>>>(nn2_w, nn2_b, root2, Wcat);
    // Layer 1 edge scatter
    edge1_kernel<<<(nEdges + 7) / 8, 256, 0, stream>>>(x, ei, ea, nn1_w, nn1_b, agg1, nEdges);
    // Dense WMMA GEMM: pqrb = relu(agg1) @ Wcat (+bias2 on base chunk)
    dim3 ggrid((nNodes + 15) / 16, 8);
    pqrb_gemm_kernel<<<ggrid, 32, 0, stream>>>(agg1, Wcat, bias2, pqrb, nNodes);
    // Layer 2 edge scatter into base chunk
    edge2_kernel<<<(nEdges + 7) / 8, 256, 0, stream>>>(ei, ea, pqrb, nEdges);
    // Head: relu -> fc1 -> relu -> fc2
    final_kernel<<<(nNodes + 255) / 256, 256, 0, stream>>>(pqrb, fc1_w, fc1_b, fc2_w, fc2_b, out, nNodes);
}